// ChamferLoss_Brute_30992484008033
// MI455X (gfx1250) — compile-verified
//
#include <hip/hip_runtime.h>
#include <math.h>

// ---------------------------------------------------------------------------
// Chamfer loss with uncertainty (sigma) weighting, B=8, M=N=3072.
// Strategy: d^2(m,n) = ||p_m||^2 + ||q_n||^2 - 2 p.q  computed as a K=4 f32
// WMMA:  A_m = [-2px,-2py,-2pz, ||p||^2],  B_n = [qx,qy,qz, 1]
//   => D = A*B = ||p||^2 - 2 p.q   (16x16 tile per v_wmma_f32_16x16x4_f32)
// VALU epilogue adds ||q||^2 and keeps running (min, argmin) per row.
// ---------------------------------------------------------------------------

typedef __attribute__((ext_vector_type(2))) float v2f;
typedef __attribute__((ext_vector_type(8))) float v8f;

#define QN_MAX 3072   // candidate-set size (M == N == 3072 in this problem)

// P: (3,Pn) query points for this batch, Q: (3,Qn) candidates.
// For each query row m: outMin[m] = min_n d^2(m,n), outIdx[m] = argmin (first).
__global__ __launch_bounds__(128) void nn_min_wmma_kernel(
    const float* __restrict__ P, const float* __restrict__ Q,
    int Pn, int Qn,
    float* __restrict__ outMin, int* __restrict__ outIdx)
{
    const int b    = blockIdx.y;
    const int tid  = threadIdx.x;
    const int lane = tid & 31;
    const int wave = tid >> 5;
    const int half = lane >> 4;     // 0: rows r..r (M=r), 1: rows M=r+8 ; K split
    const int l    = lane & 15;

    const float* __restrict__ Pb = P + (size_t)b * 3 * Pn;
    const float* __restrict__ Qb = Q + (size_t)b * 3 * Qn;

    // Stage candidates as [qx,qy,qz,||q||^2] in LDS (48KB of 320KB WGP LDS).
    __shared__ float4 q4[QN_MAX];
    for (int n = tid; n < Qn; n += blockDim.x) {
        float qx = Qb[n], qy = Qb[Qn + n], qz = Qb[2 * Qn + n];
        q4[n] = make_float4(qx, qy, qz, qx * qx + qy * qy + qz * qz);
    }
    __syncthreads();

    // Build the A tile for this wave's 16 query rows.
    // A layout (16x4 f32): lanes 0-15 hold K=0 (V0), K=1 (V1) for M=lane;
    //                      lanes 16-31 hold K=2 (V0), K=3 (V1) for M=lane-16.
    const int m0 = blockIdx.x * 64 + wave * 16;
    const int m  = m0 + l;
    const float px = Pb[m], py = Pb[Pn + m], pz = Pb[2 * Pn + m];
    const float p2 = px * px + py * py + pz * pz;
    v2f a;
    a.x = half ? (-2.0f * pz) : (-2.0f * px);
    a.y = half ? p2           : (-2.0f * py);

    float bd[8];
    int   bi[8];
#pragma unroll
    for (int r = 0; r < 8; ++r) { bd[r] = 3.4e38f; bi[r] = 0; }

    // Sweep candidate tiles of 16 columns.
    for (int n0 = 0; n0 < Qn; n0 += 16) {
        const float4 q = q4[n0 + l];          // one ds_load_b128 per lane
        // B layout (4x16 f32): V0 = K0 (lanes 0-15) / K2 (lanes 16-31),
        //                      V1 = K1 / K3 ; N = lane&15.
        v2f bt;
        bt.x = half ? q.z : q.x;
        bt.y = half ? 1.0f : q.y;

        v8f c = {};
        c = __builtin_amdgcn_wmma_f32_16x16x4_f32(
            /*neg_a=*/false, a, /*neg_b=*/false, bt,
            /*c_mod=*/(short)0, c, /*reuse_a=*/false, /*reuse_b=*/false);

        const int   n  = n0 + l;
        const float y2 = q.w;
#pragma unroll
        for (int r = 0; r < 8; ++r) {
            // c[r] holds d^2 - ||q||^2 for row M = r + 8*half, col N = l.
            float d2 = c[r] + y2;
            if (d2 < bd[r]) { bd[r] = d2; bi[r] = n; }   // strict < keeps first n
        }
    }

    // Row reduction: min over the 16 lanes of each half (lexicographic on (d2,n)
    // so ties resolve to the smallest index, matching jnp.argmin).
#pragma unroll
    for (int r = 0; r < 8; ++r) {
        float d = bd[r]; int n = bi[r];
        for (int mask = 1; mask <= 8; mask <<= 1) {
            float od = __shfl_xor(d, mask, 32);
            int   on = __shfl_xor(n, mask, 32);
            if (od < d || (od == d && on < n)) { d = od; n = on; }
        }
        bd[r] = d; bi[r] = n;
    }

    if (l == 0) {                       // lane 0 -> rows m0..m0+7, lane 16 -> +8
        const int rowBase = m0 + half * 8;
#pragma unroll
        for (int r = 0; r < 8; ++r) {
            outMin[(size_t)b * Pn + rowBase + r] = bd[r];
            outIdx[(size_t)b * Pn + rowBase + r] = bi[r];
        }
    }
}

// Single-block scalar epilogue: turns per-point (min d^2, argmin) into the 3
// scalar outputs. mean(w*d) = sum(d/sigma)/sum(1/sigma), so one pass suffices.
__global__ __launch_bounds__(256) void chamfer_finalize_kernel(
    const float* __restrict__ minsd, const int* __restrict__ idxsd,
    const float* __restrict__ minds, const int* __restrict__ idxds,
    const float* __restrict__ sigS,  const float* __restrict__ sigD,
    float* __restrict__ out, int Bn, int Mn, int Nn)
{
    const int tid = threadIdx.x;
    float s[8];
#pragma unroll
    for (int k = 0; k < 8; ++k) s[k] = 0.0f;

    for (int i = tid; i < Bn * Mn; i += 256) {          // src -> dst
        const int   b   = i / Mn;
        const float d   = sqrtf(fmaxf(minsd[i], 0.0f));
        const float sg  = 0.5f * (sigS[i] + sigD[b * Nn + idxsd[i]]);
        const float inv = 1.0f / sg;
        s[0] += logf(sg) + d * inv;
        s[1] += d;
        s[2] += inv;
        s[3] += inv * d;
    }
    for (int i = tid; i < Bn * Nn; i += 256) {          // dst -> src
        const int   b   = i / Nn;
        const float d   = sqrtf(fmaxf(minds[i], 0.0f));
        const float sg  = 0.5f * (sigD[i] + sigS[b * Mn + idxds[i]]);
        const float inv = 1.0f / sg;
        s[4] += logf(sg) + d * inv;
        s[5] += d;
        s[6] += inv;
        s[7] += inv * d;
    }

    __shared__ float red[256];
    float tot[8];
#pragma unroll
    for (int k = 0; k < 8; ++k) {
        red[tid] = s[k];
        __syncthreads();
        for (int off = 128; off > 0; off >>= 1) {
            if (tid < off) red[tid] += red[tid + off];
            __syncthreads();
        }
        tot[k] = red[0];
        __syncthreads();
    }

    if (tid == 0) {
        const float iBM = 1.0f / (float)(Bn * Mn);
        const float iBN = 1.0f / (float)(Bn * Nn);
        out[0] = tot[0] * iBM + tot[4] * iBN;           // forward + backward loss
        out[1] = tot[1] * iBM + tot[5] * iBN;           // chamfer_pure
        out[2] = tot[3] / tot[2] + tot[7] / tot[6];     // chamfer_weighted
    }
}

extern "C" void kernel_launch(void* const* d_in, const int* in_sizes, int n_in,
                              void* d_out, int out_size, void* d_ws, size_t ws_size,
                              hipStream_t stream) {
    (void)in_sizes; (void)n_in; (void)out_size; (void)ws_size;

    constexpr int B = 8, M = 3072, N = 3072;

    const float* pc_src    = (const float*)d_in[0];   // (B,3,M)
    const float* pc_dst    = (const float*)d_in[1];   // (B,3,N)
    const float* sigma_src = (const float*)d_in[2];   // (B,M)
    const float* sigma_dst = (const float*)d_in[3];   // (B,N)

    // Workspace: per-point (min d^2, argmin) for both directions.
    float* minsd = (float*)d_ws;                      // B*M
    int*   idxsd = (int*)(minsd + (size_t)B * M);     // B*M
    float* minds = (float*)(idxsd + (size_t)B * M);   // B*N
    int*   idxds = (int*)(minds + (size_t)B * N);     // B*N

    nn_min_wmma_kernel<<<dim3(M / 64, B), 128, 0, stream>>>(
        pc_src, pc_dst, M, N, minsd, idxsd);
    nn_min_wmma_kernel<<<dim3(N / 64, B), 128, 0, stream>>>(
        pc_dst, pc_src, N, M, minds, idxds);
    chamfer_finalize_kernel<<<1, 256, 0, stream>>>(
        minsd, idxsd, minds, idxds, sigma_src, sigma_dst,
        (float*)d_out, B, M, N);
}